// GNNModel_23003844838150
// MI455X (gfx1250) — compile-verified
//
#include <hip/hip_runtime.h>
#include <hip/hip_bf16.h>

#define FEAT 96                    // D
#define KSTEPS (FEAT / 4)          // 24 WMMA 16x16x4 steps over K
#define NTILES (FEAT / 16)         // 6 column tiles of 16

typedef float v2f __attribute__((ext_vector_type(2)));
typedef float v8f __attribute__((ext_vector_type(8)));

// ---------------------------------------------------------------------------
// Zero a float buffer (float4 vectorized).
// ---------------------------------------------------------------------------
__global__ __launch_bounds__(256) void zero_f32(float* __restrict__ p, long long n) {
  long long i = (long long)blockIdx.x * blockDim.x + threadIdx.x;
  long long i4 = i * 4;
  if (i4 + 3 < n) {
    *(float4*)(p + i4) = make_float4(0.f, 0.f, 0.f, 0.f);
  } else {
    for (long long j = i4; j < n; ++j) p[j] = 0.f;
  }
}

// ---------------------------------------------------------------------------
// Degree accumulation: one thread per edge.
// ---------------------------------------------------------------------------
__global__ __launch_bounds__(256) void degree_kernel(const int* __restrict__ src,
                                                     const int* __restrict__ dst,
                                                     float* __restrict__ deg_out,
                                                     float* __restrict__ deg_in,
                                                     int nedges) {
  int e = blockIdx.x * blockDim.x + threadIdx.x;
  if (e >= nedges) return;
  atomicAdd(&deg_out[src[e]], 1.0f);
  atomicAdd(&deg_in[dst[e]], 1.0f);
}

// deg -> clip(deg,1)^-0.5, in place for both arrays
__global__ __launch_bounds__(256) void norm_finalize(float* __restrict__ deg_out,
                                                     float* __restrict__ deg_in,
                                                     int n) {
  int i = blockIdx.x * blockDim.x + threadIdx.x;
  if (i >= n) return;
  deg_out[i] = 1.0f / sqrtf(fmaxf(deg_out[i], 1.0f));
  deg_in[i]  = 1.0f / sqrtf(fmaxf(deg_in[i], 1.0f));
}

// ---------------------------------------------------------------------------
// Edge scatter:  m[dst] += h[src] * norm_src[src]
// One wave per edge; edge index forced wave-uniform (readfirstlane) so the
// index/norm loads become SGPR (s_load) traffic instead of 32x-redundant VMEM.
// Lane covers features {lane, lane+32, lane+64}: 3 coalesced 128B reads +
// 96 L2-resident f32 atomics per edge (h is 19.2 MB, fits the 192 MB L2).
// ---------------------------------------------------------------------------
__global__ __launch_bounds__(256) void edge_scatter(const int* __restrict__ src,
                                                    const int* __restrict__ dst,
                                                    const float* __restrict__ h,
                                                    const float* __restrict__ norm_src,
                                                    float* __restrict__ m,
                                                    int nedges) {
  long long gid = (long long)blockIdx.x * blockDim.x + threadIdx.x;
  int e = __builtin_amdgcn_readfirstlane((int)(gid >> 5));  // wave-uniform edge id
  if (e >= nedges) return;
  int lane = threadIdx.x & 31;
  int s = src[e];   // uniform index -> scalar load
  int d = dst[e];   // uniform index -> scalar load
  float ns = norm_src[s];
  const float* __restrict__ hr = h + (long long)s * FEAT;
  float* __restrict__ mr = m + (long long)d * FEAT;
#pragma unroll
  for (int i = 0; i < 3; ++i) {
    int f = lane + i * 32;
    atomicAdd(&mr[f], hr[f] * ns);
  }
}

// ---------------------------------------------------------------------------
// Fused (m * norm_dst) @ W + b, relu  -> out     using V_WMMA_F32_16X16X4_F32
//
// Block = 256 threads = 8 waves; W (36 KB) + bias staged in LDS. Wave w owns
// one 16-row strip. All 24 A-fragments (scaled by norm_dst) are preloaded into
// 48 VGPRs. The k-loop processes ALL 6 column tiles per step with independent
// accumulators, and B-fragments for step k+1 are prefetched from LDS while the
// 6 WMMAs of step k issue -> DS latency hidden, 144 back-to-back WMMAs.
//
// A 16x4 f32 frag (2 VGPRs): lane L -> row = L&15, k = 2*(L>>4) + v
// B 4x16  f32 frag (2 VGPRs): lane L -> col = L&15, k = 2*(L>>4) + v
// C/D 16x16 f32 (8 VGPRs):    lane L, VGPR r -> row = r + 8*(L>>4), col = L&15
// All branching before/through the WMMA region is wave-uniform (EXEC all-1s).
// ---------------------------------------------------------------------------
__global__ __launch_bounds__(256) void gemm_norm_bias_relu(
    const float* __restrict__ m, const float* __restrict__ norm_dst,
    const float* __restrict__ W, const float* __restrict__ bias,
    float* __restrict__ out, int nrows) {
  __shared__ float Ws[FEAT * FEAT];
  __shared__ float bs[FEAT];

  for (int i = threadIdx.x; i < FEAT * FEAT; i += 256) Ws[i] = W[i];
  if (threadIdx.x < FEAT) bs[threadIdx.x] = bias[threadIdx.x];
  __syncthreads();

  const int wave = threadIdx.x >> 5;
  const int lane = threadIdx.x & 31;
  const int strip = blockIdx.x * 8 + wave;   // wave-uniform
  const int row0 = strip * 16;
  if (row0 >= nrows) return;                  // whole wave exits together

  const int lrow = lane & 15;
  const int lgrp = lane >> 4;

  const int arow_i = min(row0 + lrow, nrows - 1);
  const float nd = norm_dst[arow_i];
  const float* __restrict__ arow = m + (long long)arow_i * FEAT;

  v2f a[KSTEPS];
#pragma unroll
  for (int ks = 0; ks < KSTEPS; ++ks) {
    const int k = ks * 4 + 2 * lgrp;
    a[ks].x = arow[k] * nd;
    a[ks].y = arow[k + 1] * nd;
  }

  v8f acc[NTILES];
#pragma unroll
  for (int nt = 0; nt < NTILES; ++nt) acc[nt] = (v8f){};

  // B fragments for k-step 0
  v2f bf[NTILES];
#pragma unroll
  for (int nt = 0; nt < NTILES; ++nt) {
    const int k = 2 * lgrp;
    bf[nt].x = Ws[k * FEAT + nt * 16 + lrow];
    bf[nt].y = Ws[(k + 1) * FEAT + nt * 16 + lrow];
  }

#pragma unroll
  for (int ks = 0; ks < KSTEPS; ++ks) {
    v2f bn[NTILES];
    if (ks + 1 < KSTEPS) {
      const int k = (ks + 1) * 4 + 2 * lgrp;
#pragma unroll
      for (int nt = 0; nt < NTILES; ++nt) {
        bn[nt].x = Ws[k * FEAT + nt * 16 + lrow];
        bn[nt].y = Ws[(k + 1) * FEAT + nt * 16 + lrow];
      }
    }
#pragma unroll
    for (int nt = 0; nt < NTILES; ++nt) {
      acc[nt] = __builtin_amdgcn_wmma_f32_16x16x4_f32(
          /*neg_a=*/false, a[ks], /*neg_b=*/false, bf[nt],
          /*c_mod=*/(short)0, acc[nt], /*reuse_a=*/false, /*reuse_b=*/false);
    }
    if (ks + 1 < KSTEPS) {
#pragma unroll
      for (int nt = 0; nt < NTILES; ++nt) bf[nt] = bn[nt];
    }
  }

#pragma unroll
  for (int nt = 0; nt < NTILES; ++nt) {
    const int nb = nt * 16;
    const float bcol = bs[nb + lrow];
#pragma unroll
    for (int r = 0; r < 8; ++r) {
      const int row = row0 + r + 8 * lgrp;
      if (row < nrows) {
        float v = acc[nt][r] + bcol;
        out[(long long)row * FEAT + nb + lrow] = fmaxf(v, 0.0f);
      }
    }
  }
}

// ---------------------------------------------------------------------------
// Host side
// ---------------------------------------------------------------------------
extern "C" void kernel_launch(void* const* d_in, const int* in_sizes, int n_in,
                              void* d_out, int out_size, void* d_ws, size_t ws_size,
                              hipStream_t stream) {
  const float* h0  = (const float*)d_in[0];
  const int*   src = (const int*)d_in[1];
  const int*   dst = (const int*)d_in[2];
  const float* Wp[3] = {(const float*)d_in[3], (const float*)d_in[5], (const float*)d_in[7]};
  const float* bp[3] = {(const float*)d_in[4], (const float*)d_in[6], (const float*)d_in[8]};

  const int N = in_sizes[0] / FEAT;
  const int E = in_sizes[1];

  float* ws       = (float*)d_ws;
  float* norm_src = ws;                         // [N]
  float* norm_dst = ws + N;                     // [N]
  float* mbuf     = ws + 2LL * N;               // [N*FEAT]
  float* hA       = mbuf + (long long)N * FEAT; // [N*FEAT]
  float* outp     = (float*)d_out;

  const long long ND = (long long)N * FEAT;

  // degrees -> norms
  {
    long long n2 = 2LL * N;
    int blocks = (int)((n2 / 4 + 255) / 256) + 1;
    zero_f32<<<blocks, 256, 0, stream>>>(norm_src, n2);  // zeros both norm arrays
    degree_kernel<<<(E + 255) / 256, 256, 0, stream>>>(src, dst, norm_src, norm_dst, E);
    norm_finalize<<<(N + 255) / 256, 256, 0, stream>>>(norm_src, norm_dst, N);
  }

  const int strips = (N + 15) / 16;
  const int gemm_blocks = (strips + 7) / 8;
  const int zero_blocks = (int)((ND / 4 + 255) / 256) + 1;
  const long long scat_threads = (long long)E * 32;
  const int scat_blocks = (int)((scat_threads + 255) / 256);

  // ping-pong: L0 h0 -> d_out, L1 d_out -> hA, L2 hA -> d_out
  const float* lin[3]  = {h0, outp, hA};
  float*       lout[3] = {outp, hA, outp};

  for (int l = 0; l < 3; ++l) {
    zero_f32<<<zero_blocks, 256, 0, stream>>>(mbuf, ND);
    edge_scatter<<<scat_blocks, 256, 0, stream>>>(src, dst, lin[l], norm_src, mbuf, E);
    gemm_norm_bias_relu<<<gemm_blocks, 256, 0, stream>>>(mbuf, norm_dst, Wp[l], bp[l],
                                                         lout[l], N);
  }
}